// Self_Attn_6442450944418
// MI455X (gfx1250) — compile-verified
//
#include <hip/hip_runtime.h>
#include <math.h>

// Self-attention (SAGAN-style) for MI455X / gfx1250.
// B=2, C=64, CQ=8, N=F*T=8192.
// d_out = out [B,C,N] (1,048,576 f32) then attn [B,N,N] (134,217,728 f32).

#define BB 2
#define CC 64
#define CQ 8
#define NN 8192           // F*T
#define MT (NN / 16)      // 512 m-tiles per batch

typedef __attribute__((ext_vector_type(2))) float v2f;
typedef __attribute__((ext_vector_type(8))) float v8f;

// ---------------------------------------------------------------------------
// Kernel 1: 1x1-conv projections.  One thread per (b, n).
// Layouts chosen so the WMMA gathers are b64-vectorizable:
//   Q  [B,N,8]                       row-major
//   Kp [B,4,N,2]   Kp[b][o>>1][n][o&1] = K[b,o,n]   (k-row pairs adjacent)
//   Vt2[B,N/4,2,64,2]  quad q=n>>2, half h=(n>>1)&1, e=n&1:
//       Vt2[b][q][h][c][e] = V[b,c,n]               (k-pair adjacent per lane)
// ---------------------------------------------------------------------------
__global__ void proj_kernel(const float* __restrict__ x,
                            const float* __restrict__ wq, const float* __restrict__ bq,
                            const float* __restrict__ wk, const float* __restrict__ bk,
                            const float* __restrict__ wv, const float* __restrict__ bv,
                            float* __restrict__ Q, float* __restrict__ Kp,
                            float* __restrict__ Vt2) {
    const int idx = blockIdx.x * blockDim.x + threadIdx.x;   // 0 .. B*N-1
    const int b = idx / NN;
    const int n = idx - b * NN;

    const float* xb = x + (size_t)b * CC * NN + n;
    float xv[CC];
#pragma unroll
    for (int c = 0; c < CC; ++c) xv[c] = xb[(size_t)c * NN];  // lane-coalesced

#pragma unroll
    for (int o = 0; o < CQ; ++o) {
        float aq = bq[o], ak = bk[o];
#pragma unroll
        for (int c = 0; c < CC; ++c) {
            aq = fmaf(wq[o * CC + c], xv[c], aq);
            ak = fmaf(wk[o * CC + c], xv[c], ak);
        }
        Q[(size_t)idx * CQ + o] = aq;
        Kp[(((size_t)b * 4 + (o >> 1)) * NN + n) * 2 + (o & 1)] = ak;
    }

    const int q = n >> 2, h = (n >> 1) & 1, e = n & 1;
    float* vq = Vt2 + (size_t)b * NN * CC + ((size_t)q * 2 + h) * (CC * 2) + e;
#pragma unroll
    for (int o = 0; o < CC; ++o) {
        float av = bv[o];
#pragma unroll
        for (int c = 0; c < CC; ++c) av = fmaf(wv[o * CC + c], xv[c], av);
        vq[o * 2] = av;
    }
}

// ---------------------------------------------------------------------------
// Kernel 2: energy = Q @ K via V_WMMA_F32_16X16X4_F32.
// One wave per (b, m-tile, 256-column chunk).
// A (16x4 f32): row = lane%16, k = elem + 2*(lane/16)   (ISA 7.12.2)
// B (4x16 f32): col = lane%16, k = elem + 2*(lane/16)  -> one b64 from Kp
// D (16x16 f32): row = vreg + 8*(lane/16), col = lane%16
// ---------------------------------------------------------------------------
__global__ void energy_kernel(const float* __restrict__ Q,
                              const float* __restrict__ Kp,
                              float* __restrict__ attn) {
    const int wid = blockIdx.x;                 // 0 .. B*MT*32-1
    const int nchunk = wid & 31;
    const int mtile = (wid >> 5) & (MT - 1);
    const int b = wid >> 14;

    const int lane = threadIdx.x;
    const int lm = lane & 15;
    const int hi = lane >> 4;
    const int m0 = mtile * 16;

    const float* qp = Q + ((size_t)b * NN + m0 + lm) * CQ;
    v2f A0, A1;
    A0.x = qp[2 * hi];     A0.y = qp[2 * hi + 1];        // k = 0..3
    A1.x = qp[4 + 2 * hi]; A1.y = qp[5 + 2 * hi];        // k = 4..7

    const float* kp0 = Kp + ((size_t)(b * 4 + hi) * NN) * 2;       // rows 0..3
    const float* kp1 = Kp + ((size_t)(b * 4 + 2 + hi) * NN) * 2;   // rows 4..7
    float* orow = attn + ((size_t)b * NN + m0) * NN;

    for (int it = 0; it < 16; ++it) {
        const int n0 = nchunk * 256 + it * 16;
        const v2f B0 = *(const v2f*)(kp0 + (size_t)(n0 + lm) * 2);
        const v2f B1 = *(const v2f*)(kp1 + (size_t)(n0 + lm) * 2);

        v8f acc = {};
        acc = __builtin_amdgcn_wmma_f32_16x16x4_f32(false, A0, false, B0,
                                                    (short)0, acc, false, false);
        acc = __builtin_amdgcn_wmma_f32_16x16x4_f32(false, A1, false, B1,
                                                    (short)0, acc, false, false);
#pragma unroll
        for (int v = 0; v < 8; ++v)
            orow[(size_t)(v + 8 * hi) * NN + n0 + lm] = acc[v];
    }
}

// ---------------------------------------------------------------------------
// Kernel 3: row softmax fully in LDS (8192 f32 = 32KB << 320KB/WGP).
// One 256-thread block per row; exactly one global read + one write per elem.
// ---------------------------------------------------------------------------
__global__ void softmax_kernel(float* __restrict__ attn) {
    __shared__ float row[NN];
    __shared__ float red[256];
    const int tid = threadIdx.x;
    float* ptr = attn + (size_t)blockIdx.x * NN;

    for (int i = tid; i < NN; i += 256) row[i] = ptr[i];
    __syncthreads();

    float m = -INFINITY;
    for (int i = tid; i < NN; i += 256) m = fmaxf(m, row[i]);
    red[tid] = m;
    __syncthreads();
    for (int s = 128; s > 0; s >>= 1) {
        if (tid < s) red[tid] = fmaxf(red[tid], red[tid + s]);
        __syncthreads();
    }
    m = red[0];
    __syncthreads();

    float sum = 0.0f;
    for (int i = tid; i < NN; i += 256) {
        const float e = expf(row[i] - m);
        row[i] = e;
        sum += e;
    }
    red[tid] = sum;
    __syncthreads();
    for (int s = 128; s > 0; s >>= 1) {
        if (tid < s) red[tid] += red[tid + s];
        __syncthreads();
    }
    const float inv = 1.0f / red[0];

    for (int i = tid; i < NN; i += 256) ptr[i] = row[i] * inv;
}

// ---------------------------------------------------------------------------
// Kernel 4: out[b,c,m] = gamma * sum_n V[b,c,n] * attn[b,m,n] + x[b,c,m]
// One 256-thread block (8 waves) per (b, m-tile).  Each wave accumulates a
// 1/8 slice of the K(=n) range with 4 c-block WMMA accumulators, partials are
// reduced through 32KB of LDS, epilogue fuses gamma*acc + x.
// Inner iteration: 1 b64 (attn^T) + 4 b64 (Vt2) loads + 4 WMMAs.
// ---------------------------------------------------------------------------
__global__ void out_kernel(const float* __restrict__ Vt2,
                           const float* __restrict__ attn,
                           const float* __restrict__ x,
                           const float* __restrict__ gamma,
                           float* __restrict__ out) {
    __shared__ float part[8 * 1024];            // 8 waves x (4 cb x 8 v x 32 lanes)

    const int wid = blockIdx.x;                 // 0 .. B*MT-1
    const int b = wid / MT;
    const int mtile = wid - b * MT;
    const int m0 = mtile * 16;

    const int tid = threadIdx.x;
    const int wave = tid >> 5;                  // 0..7
    const int lane = tid & 31;
    const int lm = lane & 15;
    const int hi = lane >> 4;

    const float* aRow = attn + ((size_t)b * NN + m0 + lm) * NN;
    const float* vBase = Vt2 + (size_t)b * NN * CC;
    const int nbase = wave * (NN / 8);          // 1024 n per wave

    v8f acc0 = {}, acc1 = {}, acc2 = {}, acc3 = {};

    for (int i = 0; i < NN / 8 / 4; ++i) {      // 256 iterations
        const int n0 = nbase + i * 4;
        // B-tile: k = elem + 2*hi, col = lm  ->  attn[m0+lm][n0 + k]
        const v2f Bt = *(const v2f*)(aRow + n0 + 2 * hi);
        // A-tiles: row c = cb*16+lm, k = elem + 2*hi -> Vt2 quad layout
        const float* vq = vBase + ((size_t)(n0 >> 2) * 2 + hi) * (CC * 2) + lm * 2;
        const v2f A0 = *(const v2f*)(vq);
        const v2f A1 = *(const v2f*)(vq + 32);
        const v2f A2 = *(const v2f*)(vq + 64);
        const v2f A3 = *(const v2f*)(vq + 96);

        acc0 = __builtin_amdgcn_wmma_f32_16x16x4_f32(false, A0, false, Bt,
                                                     (short)0, acc0, false, false);
        acc1 = __builtin_amdgcn_wmma_f32_16x16x4_f32(false, A1, false, Bt,
                                                     (short)0, acc1, false, false);
        acc2 = __builtin_amdgcn_wmma_f32_16x16x4_f32(false, A2, false, Bt,
                                                     (short)0, acc2, false, false);
        acc3 = __builtin_amdgcn_wmma_f32_16x16x4_f32(false, A3, false, Bt,
                                                     (short)0, acc3, false, false);
    }

    // Dump per-wave partials: part[wave][cb][v][lane]
    float* pw = part + wave * 1024 + lane;
#pragma unroll
    for (int v = 0; v < 8; ++v) {
        pw[(0 * 8 + v) * 32] = acc0[v];
        pw[(1 * 8 + v) * 32] = acc1[v];
        pw[(2 * 8 + v) * 32] = acc2[v];
        pw[(3 * 8 + v) * 32] = acc3[v];
    }
    __syncthreads();

    // Cross-wave reduction + epilogue: 1024 outputs, 4 per thread.
    const float g = gamma[0];
#pragma unroll
    for (int r = 0; r < 4; ++r) {
        const int o = tid + r * 256;            // ((cb*8 + v)*32 + lane)
        float s = 0.0f;
#pragma unroll
        for (int w = 0; w < 8; ++w) s += part[w * 1024 + o];

        const int ol = o & 31;
        const int ov = (o >> 5) & 7;
        const int ocb = o >> 8;
        const int c = ocb * 16 + ov + 8 * (ol >> 4);
        const size_t idx = ((size_t)b * CC + c) * NN + m0 + (ol & 15);
        out[idx] = fmaf(g, s, x[idx]);
    }
}

// ---------------------------------------------------------------------------
extern "C" void kernel_launch(void* const* d_in, const int* in_sizes, int n_in,
                              void* d_out, int out_size, void* d_ws, size_t ws_size,
                              hipStream_t stream) {
    const float* x     = (const float*)d_in[0];
    const float* wq    = (const float*)d_in[1];
    const float* bq    = (const float*)d_in[2];
    const float* wk    = (const float*)d_in[3];
    const float* bk    = (const float*)d_in[4];
    const float* wv    = (const float*)d_in[5];
    const float* bv    = (const float*)d_in[6];
    const float* gamma = (const float*)d_in[7];

    float* out  = (float*)d_out;                       // [B,C,N]
    float* attn = out + (size_t)BB * CC * NN;          // [B,N,N]

    float* Q   = (float*)d_ws;                         // [B,N,8]    131072 f
    float* Kp  = Q + (size_t)BB * NN * CQ;             // [B,4,N,2]  131072 f
    float* Vt2 = Kp + (size_t)BB * CQ * NN;            // [B,N,64]  1048576 f

    proj_kernel<<<(BB * NN) / 256, 256, 0, stream>>>(x, wq, bq, wk, bk, wv, bv,
                                                     Q, Kp, Vt2);
    energy_kernel<<<BB * MT * 32, 32, 0, stream>>>(Q, Kp, attn);
    softmax_kernel<<<BB * NN, 256, 0, stream>>>(attn);
    out_kernel<<<BB * MT, 256, 0, stream>>>(Vt2, attn, x, gamma, out);
}